// IMUHOIModel_32066225832133
// MI455X (gfx1250) — compile-verified
//
#include <hip/hip_runtime.h>
#include <stdint.h>

// Problem constants from the reference
#define BB 512
#define SS 4096
#define TS 8                 // time steps per prefetch tile
#define NT (SS / TS)         // 512 tiles
// LDS per-lane strides (floats), padded to stay 16B-aligned and reduce bank conflicts
#define CBS (2 * TS + 4)     // 20 dwords (contact: 2 floats/step)
#define HBS (6 * TS + 4)     // 52 dwords (hands:   6 floats/step)
#define RBS (9 * TS + 4)     // 76 dwords (rotm:    9 floats/step)

#define C_LOADS ((2 * TS * 4) / 16)   // 4  b128 per lane per tile
#define H_LOADS ((6 * TS * 4) / 16)   // 12 b128
#define R_LOADS ((9 * TS * 4) / 16)   // 18 b128
#define TILE_LOADS (C_LOADS + H_LOADS + R_LOADS)  // 34 (< 63, fits 6-bit ASYNCcnt)

// Generic shared pointer -> 32-bit LDS offset (low 32 bits of the LDS aperture address)
__device__ __forceinline__ unsigned lds_off(const void* p) {
  return (unsigned)(uintptr_t)p;
}

// CDNA5 async copy: copies 16 bytes from global base plus byte offset goff to the LDS address in lds, tracked by ASYNCcnt.
// GVS addressing: uniform 64-bit SGPR base + per-lane 32-bit VGPR byte offset.
__device__ __forceinline__ void async_b128(unsigned lds, const float* base, unsigned goff) {
  asm volatile("global_load_async_to_lds_b128 %0, %1, %2"
               :: "v"(lds), "v"(goff), "s"(base) : "memory");
}

__global__ __launch_bounds__(32) void fk_obj_trans_kernel(
    const float* __restrict__ prob,    // (B,S,2)
    const float* __restrict__ hand,    // (B,S,2,3)
    const float* __restrict__ rotm,    // (B,S,3,3)
    const float* __restrict__ tinit,   // (B,3)
    float* __restrict__ out)           // (B,S,3)
{
  __shared__ float cbuf[2][32][CBS];
  __shared__ float hbuf[2][32][HBS];
  __shared__ float rbuf[2][32][RBS];

  const int lane = threadIdx.x;
  const int b    = blockIdx.x * 32 + lane;

  // Row base byte offsets (max ~75 MB < 2^31, so 32-bit offsets are safe)
  const unsigned crow = (unsigned)b * (unsigned)(SS * 2 * 4);
  const unsigned hrow = (unsigned)b * (unsigned)(SS * 6 * 4);
  const unsigned rrow = (unsigned)b * (unsigned)(SS * 9 * 4);

  const unsigned c_lds[2] = { lds_off(&cbuf[0][lane][0]), lds_off(&cbuf[1][lane][0]) };
  const unsigned h_lds[2] = { lds_off(&hbuf[0][lane][0]), lds_off(&hbuf[1][lane][0]) };
  const unsigned r_lds[2] = { lds_off(&rbuf[0][lane][0]), lds_off(&rbuf[1][lane][0]) };

  auto issue_tile = [&](int k, int bi) {
    const unsigned co = crow + (unsigned)k * (TS * 2 * 4);
    const unsigned ho = hrow + (unsigned)k * (TS * 6 * 4);
    const unsigned ro = rrow + (unsigned)k * (TS * 9 * 4);
#pragma unroll
    for (int j = 0; j < C_LOADS; ++j) async_b128(c_lds[bi] + j * 16, prob, co + j * 16);
#pragma unroll
    for (int j = 0; j < H_LOADS; ++j) async_b128(h_lds[bi] + j * 16, hand, ho + j * 16);
#pragma unroll
    for (int j = 0; j < R_LOADS; ++j) async_b128(r_lds[bi] + j * 16, rotm, ro + j * 16);
  };

  // Scan carry state
  int   cur  = -1;
  float ox = tinit[b * 3 + 0], oy = tinit[b * 3 + 1], oz = tinit[b * 3 + 2];
  float dlx = 0.f, dly = 0.f, dlz = 0.f;
  float dist = 0.1f;
  bool  plc = false, prc = false;

  float* op = out + (size_t)b * (size_t)(SS * 3);

  issue_tile(0, 0);

  for (int k = 0; k < NT; ++k) {
    const int bi = k & 1;
    if (k + 1 < NT) {
      // Make sure last tile's LDS reads have drained before the async engine
      // overwrites that buffer, then prefetch tile k+1 into the other buffer.
      asm volatile("s_wait_dscnt 0" ::: "memory");
      issue_tile(k + 1, bi ^ 1);
      // Tile k complete when only tile (k+1)'s TILE_LOADS remain outstanding.
      asm volatile("s_wait_asynccnt 34" ::: "memory");
    } else {
      asm volatile("s_wait_asynccnt 0" ::: "memory");
    }

#pragma unroll
    for (int j = 0; j < TS; ++j) {
      const float p0 = cbuf[bi][lane][2 * j + 0];
      const float p1 = cbuf[bi][lane][2 * j + 1];
      const float* H = &hbuf[bi][lane][6 * j];
      const float* R = &rbuf[bi][lane][9 * j];

      const bool first = (k == 0) && (j == 0);
      const bool lc = (p0 > 0.5f) && !first;   // l_c with t=0 forced false
      const bool rc = (p1 > 0.5f) && !first;
      const bool ls = lc && !plc;              // contact-start events
      const bool rs = rc && !prc;
      plc = lc; prc = rc;

      const bool has_c = (cur == 0) ? lc : ((cur == 1) ? rc : false);
      const bool has_o = (cur == 0) ? rc : ((cur == 1) ? lc : false);
      const bool grab  = ls || rs;
      const bool sw    = !grab && !has_c && has_o && (cur != -1);
      const bool rel   = !grab && !sw && !has_c && (cur != -1);
      const int  nw    = ls ? 0 : 1;           // valid only when grab
      const int  cur2  = grab ? nw : (sw ? (cur ^ 1) : (rel ? -1 : cur));

      // hand_sel = hand[max(cur2,0)] : load both, branchless select
      const float h0x = H[0], h0y = H[1], h0z = H[2];
      const float h1x = H[3], h1y = H[4], h1z = H[5];
      const bool  i1  = cur2 > 0;
      const float hx = i1 ? h1x : h0x;
      const float hy = i1 ? h1y : h0y;
      const float hz = i1 ? h1z : h0z;

      const bool regrab = grab || sw;
      const float vx = ox - hx, vy = oy - hy, vz = oz - hz;
      const float s2 = vx * vx + vy * vy + vz * vz;
      const bool  safe = s2 > 1e-12f;
      const float s2m = safe ? s2 : 1.0f;

      // norm via v_rsq_f32 + one Newton-Raphson step (~1 ulp), replacing the
      // correctly-rounded sqrt + full div sequences: shortest serial chain.
      float r = __builtin_amdgcn_rsqf(s2m);
      r = r * __builtin_fmaf(-0.5f * s2m, r * r, 1.5f);
      const float ux = safe ? (vx * r) : 0.0f;
      const float uy = safe ? (vy * r) : 0.0f;
      const float uz = safe ? (vz * r) : 1.0f;
      const float nd = safe ? (s2m * r) : 0.1f;   // s2 * rsqrt(s2) == sqrt(s2)

      // dir_new = R^T @ unit
      const float dnx = __builtin_fmaf(R[0], ux, __builtin_fmaf(R[3], uy, R[6] * uz));
      const float dny = __builtin_fmaf(R[1], ux, __builtin_fmaf(R[4], uy, R[7] * uz));
      const float dnz = __builtin_fmaf(R[2], ux, __builtin_fmaf(R[5], uy, R[8] * uz));
      dlx = regrab ? dnx : dlx;
      dly = regrab ? dny : dly;
      dlz = regrab ? dnz : dlz;
      dist = regrab ? nd : dist;

      // obj_new = hand + (R @ dir_local) * dist
      const bool att = (cur2 != -1);
      const float mx = __builtin_fmaf(R[0], dlx, __builtin_fmaf(R[1], dly, R[2] * dlz));
      const float my = __builtin_fmaf(R[3], dlx, __builtin_fmaf(R[4], dly, R[5] * dlz));
      const float mz = __builtin_fmaf(R[6], dlx, __builtin_fmaf(R[7], dly, R[8] * dlz));
      const float wx = __builtin_fmaf(mx, dist, hx);
      const float wy = __builtin_fmaf(my, dist, hy);
      const float wz = __builtin_fmaf(mz, dist, hz);
      ox = att ? wx : ox;
      oy = att ? wy : oy;
      oz = att ? wz : oz;
      cur = cur2;

      const int t = k * TS + j;
      op[t * 3 + 0] = ox;
      op[t * 3 + 1] = oy;
      op[t * 3 + 2] = oz;
    }
  }
}

extern "C" void kernel_launch(void* const* d_in, const int* in_sizes, int n_in,
                              void* d_out, int out_size, void* d_ws, size_t ws_size,
                              hipStream_t stream) {
  (void)in_sizes; (void)n_in; (void)d_ws; (void)ws_size; (void)out_size;
  const float* prob  = (const float*)d_in[0];  // (512,4096,2)
  const float* hand  = (const float*)d_in[1];  // (512,4096,2,3)
  const float* rotm  = (const float*)d_in[2];  // (512,4096,3,3)
  const float* tinit = (const float*)d_in[3];  // (512,3)
  float* out = (float*)d_out;                  // (512,4096,3)
  fk_obj_trans_kernel<<<BB / 32, 32, 0, stream>>>(prob, hand, rotm, tinit, out);
}